// SpecialSpmmFinal_3229815406782
// MI455X (gfx1250) — compile-verified
//
#include <hip/hip_runtime.h>
#include <stdint.h>

// Segment-sum (scatter-add) for MI455X / gfx1250.
//   out[r, :] += edge_w[e, :]  for r = edge[0][e]
// N=100000, E=1250000, F=64 f32.  HBM-bound: ~330MB -> ~15us floor @ 23.3TB/s.
//
// CDNA5 path: Tensor Data Mover streams edge_w tiles into LDS (TENSORcnt,
// double-buffered), waves drain tiles with hardware fp32 global atomics
// (output is 25.6MB -> L2-resident RMW).

#define F_FEATS       64
#define CHUNK_EDGES   96
#define CHUNK_ELEMS   (CHUNK_EDGES * F_FEATS)        // 6144 f32 = 24KB
#define CHUNK_F4      (CHUNK_ELEMS / 4)              // 1536 float4
#define BLOCK_THREADS 256
#define F4_PER_THREAD (CHUNK_F4 / BLOCK_THREADS)     // 6

typedef unsigned int v4u __attribute__((ext_vector_type(4)));
typedef int          v8i __attribute__((ext_vector_type(8)));
typedef int          v4i __attribute__((ext_vector_type(4)));

// ---------------------------------------------------------------------------
// Kernel 1: zero the output (harness poisons d_out with 0xAA).
// ---------------------------------------------------------------------------
__global__ void zero_out_kernel(float* __restrict__ out, int n) {
    int i = blockIdx.x * blockDim.x + threadIdx.x;
    int base = i * 4;
    if (base + 4 <= n) {
        reinterpret_cast<float4*>(out)[i] = make_float4(0.f, 0.f, 0.f, 0.f);
    } else {
        for (int k = base; k < n; ++k) out[k] = 0.f;
    }
}

// ---------------------------------------------------------------------------
// TDM: issue a 1-D tile load of CHUNK_ELEMS f32 from `gsrc` into LDS offset
// `lds_off`.  tensor_dim0 = remaining elements, so the tail chunk's OOB
// reads return zero (never consumed).  Descriptor per CDNA5 ISA sec. 8.
// ---------------------------------------------------------------------------
__device__ __forceinline__ void tdm_load_chunk(const float* gsrc,
                                               uint32_t lds_off,
                                               uint32_t rem_elems) {
    uint64_t ga = (uint64_t)(uintptr_t)gsrc;

    // D# group 0 (128b): count=1 | lds_addr | global_addr[56:0] | type=2
    v4u g0;
    g0[0] = 1u;                                           // count=1, user desc
    g0[1] = lds_off;                                      // LDS byte address
    g0[2] = (uint32_t)ga;                                 // global_addr[31:0]
    g0[3] = (uint32_t)((ga >> 32) & 0x01FFFFFFu)          // global_addr[56:32]
            | (2u << 30);                                 // type=2 ("image")

    // D# group 1 (256b): data_size=4B, 1-D tile.
    //  bits 17:16      data_size = 2 (4 bytes)
    //  bits 79:48      tensor_dim0 = rem_elems
    //  bits 111:80     tensor_dim1 = 1
    //  bits 127:112    tile_dim0   = CHUNK_ELEMS
    //  bits 143:128    tile_dim1   = 0 (unused -> 1-D)
    v8i g1;
    g1[0] = (int)(2u << 16);                              // data_size
    g1[1] = (int)((rem_elems & 0xFFFFu) << 16);           // tensor_dim0[15:0]
    g1[2] = (int)((rem_elems >> 16) | (1u << 16));        // dim0[31:16], dim1=1
    g1[3] = (int)((uint32_t)CHUNK_ELEMS << 16);           // tile_dim0
    g1[4] = 0;                                            // tile_dim1/2 = 0
    g1[5] = (int)rem_elems;                               // dim0_stride (unused 1-D)
    g1[6] = 0;
    g1[7] = 0;

    v4i z = {0, 0, 0, 0};                                 // groups 2/3 unused
#if defined(__clang_major__) && (__clang_major__ >= 23)
    v8i z8 = {0, 0, 0, 0, 0, 0, 0, 0};
    __builtin_amdgcn_tensor_load_to_lds(g0, g1, z, z, z8, 0);
#else
    __builtin_amdgcn_tensor_load_to_lds(g0, g1, z, z, 0);
#endif
}

// ---------------------------------------------------------------------------
// Kernel 2: TDM-staged scatter-add.
// Each workgroup owns a contiguous span of chunks; wave 0 double-buffers
// chunk tiles into LDS via TDM; all 8 waves drain a tile with ds_load_b128
// + 4x global_atomic_add_f32 per float4.
// ---------------------------------------------------------------------------
__global__ void __launch_bounds__(BLOCK_THREADS)
segsum_tdm_kernel(const long long* __restrict__ rows,
                  const float* __restrict__ w,
                  float* __restrict__ out,
                  int E, int chunks_per_wg, int total_chunks) {
    __shared__ __align__(16) float buf[2][CHUNK_ELEMS];   // 48KB

    const int start = blockIdx.x * chunks_per_wg;
    int nmine = total_chunks - start;
    if (nmine > chunks_per_wg) nmine = chunks_per_wg;
    if (nmine <= 0) return;                               // uniform per WG

    const int tid = threadIdx.x;
    const bool issuer = (tid == 0);                       // one wave issues TDM
    const long long total_elems = (long long)E * F_FEATS;

    // Prologue: kick off chunk 0 into buf[0].
    if (issuer) {
        long long base_el = (long long)start * CHUNK_ELEMS;
        tdm_load_chunk(w + base_el,
                       (uint32_t)(uintptr_t)&buf[0][0],
                       (uint32_t)(total_elems - base_el));
    }

    for (int it = 0; it < nmine; ++it) {
        const int cb = it & 1;

        if (issuer) {
            if (it + 1 < nmine) {
                // Issue next tile into the other buffer, then wait for the
                // current one (tensor ops complete in order per wave).
                long long base_el = (long long)(start + it + 1) * CHUNK_ELEMS;
                tdm_load_chunk(w + base_el,
                               (uint32_t)(uintptr_t)&buf[cb ^ 1][0],
                               (uint32_t)(total_elems - base_el));
                __builtin_amdgcn_s_wait_tensorcnt(1);
            } else {
                __builtin_amdgcn_s_wait_tensorcnt(0);
            }
        }
        __syncthreads();                                  // tile visible to all

        const int edge_base = (start + it) * CHUNK_EDGES;
        int chunk_edges = E - edge_base;
        if (chunk_edges > CHUNK_EDGES) chunk_edges = CHUNK_EDGES;

        // Keep the index stream ahead (global_prefetch_b8, speculative).
        __builtin_prefetch(rows + edge_base + CHUNK_EDGES + tid, 0, 0);

        #pragma unroll
        for (int k = 0; k < F4_PER_THREAD; ++k) {
            const int u  = tid + k * BLOCK_THREADS;       // float4 slot in tile
            const int el = u >> 4;                        // local edge
            const int c  = u & 15;                        // float4 chunk in row
            if (el < chunk_edges) {
                long long r = rows[edge_base + el];       // edge[0][e] (int64)
                float4 v = *reinterpret_cast<const float4*>(&buf[cb][u << 2]);
                float* dst = out + r * F_FEATS + (c << 2);
                unsafeAtomicAdd(dst + 0, v.x);
                unsafeAtomicAdd(dst + 1, v.y);
                unsafeAtomicAdd(dst + 2, v.z);
                unsafeAtomicAdd(dst + 3, v.w);
            }
        }
        __syncthreads();                                  // buffer reuse fence
    }
}

// ---------------------------------------------------------------------------
// Launch. Inputs (setup_inputs order): edge [2,E] int64, edge_w [E,F] f32,
// then scalar arrays N, E, out_features (derived from sizes instead).
// ---------------------------------------------------------------------------
extern "C" void kernel_launch(void* const* d_in, const int* in_sizes, int n_in,
                              void* d_out, int out_size, void* d_ws, size_t ws_size,
                              hipStream_t stream) {
    const long long* edge   = (const long long*)d_in[0];  // row idx = first E
    const float*     edge_w = (const float*)d_in[1];
    float*           out    = (float*)d_out;

    const int E = in_sizes[0] / 2;                        // edge is [2, E]

    // Zero output.
    {
        int n4 = (out_size + 3) / 4;
        int blocks = (n4 + 255) / 256;
        zero_out_kernel<<<blocks, 256, 0, stream>>>(out, out_size);
    }

    // TDM-staged scatter-add.
    {
        const int total_chunks = (E + CHUNK_EDGES - 1) / CHUNK_EDGES;  // 13021
        const int target_wgs   = 4096;
        int cpw = (total_chunks + target_wgs - 1) / target_wgs;        // ~4
        if (cpw < 1) cpw = 1;
        int blocks = (total_chunks + cpw - 1) / cpw;
        segsum_tdm_kernel<<<blocks, BLOCK_THREADS, 0, stream>>>(
            edge, edge_w, out, E, cpw, total_chunks);
    }
}